// STGCN_37434934952459
// MI455X (gfx1250) — compile-verified
//
#include <hip/hip_runtime.h>
#include <math.h>

#define TT 8
#define NN 50000
#define EE 600000
#define CH 64

typedef __attribute__((ext_vector_type(2))) float v2f;
typedef __attribute__((ext_vector_type(4))) float v4f;
typedef __attribute__((ext_vector_type(8))) float v8f;

__device__ __forceinline__ v8f wmma_f32(v2f a, v2f b, v8f c) {
  // D = A(16x4,f32) * B(4x16,f32) + C(16x16,f32)
  return __builtin_amdgcn_wmma_f32_16x16x4_f32(false, a, false, b, (short)0, c, false, false);
}

// Swizzled B-tile LDS index: element B[k][n] lives at (k>>2)*4*NCOL + n*4 + (k&3).
// A lane's (B[k0][col], B[k0+1][col]) pair (k0 even) is then contiguous ->
// single aligned ds_load_b64 straight into the WMMA operand VGPR pair, and the
// 32 lanes of a wave touch all 64 LDS banks exactly once (conflict-free).

// ---------------------------------------------------------------------------
// Edge preprocessing: softmax over edge weights, degree, symmetric norm
// ---------------------------------------------------------------------------
__global__ void k_init_scalar(int* maxbits, float* sumv) {
  if (threadIdx.x == 0) { *maxbits = (int)0xFF800000u; *sumv = 0.f; }
}

__global__ void k_init_deg(float* deg) {
  int i = blockIdx.x * 256 + threadIdx.x;
  if (i < NN) deg[i] = 1.0f;  // self-loop weight
}

__global__ void k_edge_max(const float* __restrict__ w, int* maxbits) {
  __shared__ float red[256];
  int i = blockIdx.x * 256 + threadIdx.x;
  red[threadIdx.x] = (i < EE) ? w[i] : -1e30f;
  __syncthreads();
  for (int s = 128; s > 0; s >>= 1) {
    if (threadIdx.x < s) red[threadIdx.x] = fmaxf(red[threadIdx.x], red[threadIdx.x + s]);
    __syncthreads();
  }
  if (threadIdx.x == 0) atomicMax(maxbits, __float_as_int(red[0]));  // weights >= 0
}

__global__ void k_edge_expsum(const float* __restrict__ w, const int* maxbits, float* sumv) {
  __shared__ float red[256];
  float mx = __int_as_float(*maxbits);
  int i = blockIdx.x * 256 + threadIdx.x;
  red[threadIdx.x] = (i < EE) ? expf(w[i] - mx) : 0.f;
  __syncthreads();
  for (int s = 128; s > 0; s >>= 1) {
    if (threadIdx.x < s) red[threadIdx.x] += red[threadIdx.x + s];
    __syncthreads();
  }
  if (threadIdx.x == 0) unsafeAtomicAdd(sumv, red[0]);
}

__global__ void k_edge_wsm(const float* __restrict__ w, const long long* __restrict__ ei,
                           const int* maxbits, const float* sumv,
                           float* __restrict__ wsm, float* deg) {
  int e = blockIdx.x * 256 + threadIdx.x;
  if (e >= EE) return;
  float s = expf(w[e] - __int_as_float(*maxbits)) / (*sumv);
  wsm[e] = s;
  long long col = ei[(size_t)EE + e];
  unsafeAtomicAdd(deg + col, s);
}

__global__ void k_dinv(const float* __restrict__ deg, float* __restrict__ dinv,
                       float* __restrict__ selfn) {
  int i = blockIdx.x * 256 + threadIdx.x;
  if (i < NN) {
    float d = deg[i];
    float r = (d > 0.f) ? rsqrtf(d) : 0.f;
    dinv[i] = r;
    selfn[i] = r * r;
  }
}

__global__ void k_norm(const long long* __restrict__ ei, const float* __restrict__ wsm,
                       const float* __restrict__ dinv, float* __restrict__ nrm) {
  int e = blockIdx.x * 256 + threadIdx.x;
  if (e >= EE) return;
  long long r = ei[e], c = ei[(size_t)EE + e];
  nrm[e] = dinv[r] * wsm[e] * dinv[c];
}

// ---------------------------------------------------------------------------
// Fused temporal-conv (3 shifted WMMA GEMMs) + clip + InstanceNorm(T) + ReLU.
// Output layout [N][T][64], rows r = n*8+t. 256 thr = 8 waves; each wave owns
// a 16-row x 64-col tile = exactly 2 nodes x 8 timesteps, so the f32 D-fragment
// layout makes InstanceNorm over T lane-local across the 8 accumulator VGPRs.
// ---------------------------------------------------------------------------
template <int CIN, bool TNC>
__global__ void __launch_bounds__(256)
k_conv_in(const float* __restrict__ xin, const float* __restrict__ tw,
          const float* __restrict__ tb, const float* __restrict__ gam,
          const float* __restrict__ bet, float* __restrict__ hout) {
  __shared__ float sB[3 * CIN * 64];  // swizzled per dt
  for (int i = threadIdx.x; i < 3 * CIN * 64; i += 256) {
    int co = i & 63;
    int ci = (i >> 6) % CIN;
    int dt = i / (CIN * 64);
    // B_dt[ci][co] = w[co][ci][dt], swizzled
    sB[dt * (CIN * 64) + ((ci >> 2) * 256) + co * 4 + (ci & 3)] =
        tw[(co * CIN + ci) * 3 + dt];
  }
  __syncthreads();

  const int wave = threadIdx.x >> 5;
  const int lane = threadIdx.x & 31;
  const int half = lane >> 4;
  const int l16 = lane & 15;
  const int rowBase = blockIdx.x * 128 + wave * 16;

  const int r = rowBase + l16;  // A-fragment row for this lane
  const int n = r >> 3;
  const int t = r & 7;

  v8f acc[4];
  float g4[4], b4[4];
#pragma unroll
  for (int nt = 0; nt < 4; ++nt) {
    int c = nt * 16 + l16;
    float bia = tb[c];
    g4[nt] = gam[c];
    b4[nt] = bet[c];
#pragma unroll
    for (int i = 0; i < 8; ++i) acc[nt][i] = bia;
  }

  for (int kb = 0; kb < CIN / 4; ++kb) {
    const int k0 = kb * 4 + half * 2;
    v2f a[3];
#pragma unroll
    for (int dt = 0; dt < 3; ++dt) {
      int ts = t + dt - 1;
      v2f av; av.x = 0.f; av.y = 0.f;
      if (ts >= 0 && ts < 8) {
        const float* p = TNC ? (xin + ((size_t)ts * NN + n) * CIN + k0)
                             : (xin + ((size_t)n * 8 + ts) * CIN + k0);
        av = *(const v2f*)p;
      }
      a[dt] = av;
    }
#pragma unroll
    for (int nt = 0; nt < 4; ++nt) {
#pragma unroll
      for (int dt = 0; dt < 3; ++dt) {
        v2f b = *(const v2f*)&sB[dt * (CIN * 64) + kb * 256 +
                                 (nt * 16 + l16) * 4 + half * 2];
        acc[nt] = wmma_f32(a[dt], b, acc[nt]);
      }
    }
  }

  // clip -> InstanceNorm over T (lane-local across 8 VGPRs) -> affine -> ReLU
  const int drow0 = rowBase + half * 8;
#pragma unroll
  for (int nt = 0; nt < 4; ++nt) {
    float m = 0.f;
#pragma unroll
    for (int i = 0; i < 8; ++i) {
      float v = fminf(fmaxf(acc[nt][i], -10.f), 10.f);
      acc[nt][i] = v;
      m += v;
    }
    m *= 0.125f;
    float s2 = 0.f;
#pragma unroll
    for (int i = 0; i < 8; ++i) { float d = acc[nt][i] - m; s2 += d * d; }
    float rstd = rsqrtf(s2 * 0.125f + 1e-5f);
    int c = nt * 16 + l16;
#pragma unroll
    for (int i = 0; i < 8; ++i) {
      float y = (acc[nt][i] - m) * rstd * g4[nt] + b4[nt];
      hout[(size_t)(drow0 + i) * 64 + c] = fmaxf(y, 0.f);
    }
  }
}

// ---------------------------------------------------------------------------
// hw = h @ gw : [400000 x 64] x [64 x 64], fp32 WMMA
// ---------------------------------------------------------------------------
__global__ void __launch_bounds__(256)
k_gemm64(const float* __restrict__ A, const float* __restrict__ W, float* __restrict__ O) {
  __shared__ float sB[64 * 64];  // swizzled
  for (int i = threadIdx.x; i < 64 * 64; i += 256) {
    int k = i >> 6, nn = i & 63;
    sB[((k >> 2) * 256) + nn * 4 + (k & 3)] = W[i];
  }
  __syncthreads();
  const int wave = threadIdx.x >> 5;
  const int lane = threadIdx.x & 31;
  const int half = lane >> 4;
  const int l16 = lane & 15;
  const int rowBase = blockIdx.x * 128 + wave * 16;

  v8f acc[4];
#pragma unroll
  for (int nt = 0; nt < 4; ++nt)
#pragma unroll
    for (int i = 0; i < 8; ++i) acc[nt][i] = 0.f;

  const float* arow = A + (size_t)(rowBase + l16) * 64;
  for (int kb = 0; kb < 16; ++kb) {
    const int k0 = kb * 4 + half * 2;
    v2f a = *(const v2f*)(arow + k0);
#pragma unroll
    for (int nt = 0; nt < 4; ++nt) {
      v2f b = *(const v2f*)&sB[kb * 256 + (nt * 16 + l16) * 4 + half * 2];
      acc[nt] = wmma_f32(a, b, acc[nt]);
    }
  }
  const int drow0 = rowBase + half * 8;
#pragma unroll
  for (int nt = 0; nt < 4; ++nt)
#pragma unroll
    for (int i = 0; i < 8; ++i)
      O[(size_t)(drow0 + i) * 64 + nt * 16 + l16] = acc[nt][i];
}

// ---------------------------------------------------------------------------
// Graph aggregation: self-loop init, per-edge scatter (2KB contiguous), finalize
// ---------------------------------------------------------------------------
__global__ void k_selfloop(const v4f* __restrict__ hw4, const float* __restrict__ selfn,
                           v4f* __restrict__ agg4) {
  size_t i = (size_t)blockIdx.x * 256 + threadIdx.x;  // over N*T*64/4 float4s
  if (i < (size_t)NN * 128) {
    float s = selfn[i >> 7];  // 128 float4s per node
    v4f v = hw4[i];
    v4f o; o.x = s * v.x; o.y = s * v.y; o.z = s * v.z; o.w = s * v.w;
    agg4[i] = o;
  }
}

__global__ void __launch_bounds__(256)
k_scatter(const float* __restrict__ hw, const long long* __restrict__ ei,
          const float* __restrict__ nrm, float* __restrict__ agg) {
  const int e = blockIdx.x;
  const long long r = ei[e];
  const long long c = ei[(size_t)EE + e];
  const float nm = nrm[e];
  const float* src = hw + (size_t)r * 512;  // node payload: 8 T x 64 ch contiguous
  float* dst = agg + (size_t)c * 512;
#pragma unroll
  for (int j = threadIdx.x; j < 512; j += 256)
    unsafeAtomicAdd(dst + j, nm * src[j]);
}

__global__ void k_finalize(v4f* __restrict__ agg4, const v4f* __restrict__ gb4) {
  size_t i = (size_t)blockIdx.x * 256 + threadIdx.x;  // over N*T*64/4 float4s
  if (i < (size_t)NN * 128) {
    v4f v = agg4[i];
    v4f b = gb4[i & 15];  // 16 float4s cover 64 channels
    v4f o;
    o.x = fminf(fmaxf(v.x + b.x, 0.f), 10.f);
    o.y = fminf(fmaxf(v.y + b.y, 0.f), 10.f);
    o.z = fminf(fmaxf(v.z + b.z, 0.f), 10.f);
    o.w = fminf(fmaxf(v.w + b.w, 0.f), 10.f);
    agg4[i] = o;  // clip(relu(x),-10,10)
  }
}

// ---------------------------------------------------------------------------
// Head: [400000 x 64] x [64 x 16] WMMA, then mean over T + bias
// ---------------------------------------------------------------------------
__global__ void __launch_bounds__(256)
k_headgemm(const float* __restrict__ A, const float* __restrict__ W, float* __restrict__ O) {
  __shared__ float sB[64 * 16];  // swizzled
  for (int i = threadIdx.x; i < 64 * 16; i += 256) {
    int k = i >> 4, nn = i & 15;
    sB[((k >> 2) * 64) + nn * 4 + (k & 3)] = W[i];
  }
  __syncthreads();
  const int wave = threadIdx.x >> 5;
  const int lane = threadIdx.x & 31;
  const int half = lane >> 4;
  const int l16 = lane & 15;
  const int rowBase = blockIdx.x * 128 + wave * 16;

  v8f acc;
#pragma unroll
  for (int i = 0; i < 8; ++i) acc[i] = 0.f;

  const float* arow = A + (size_t)(rowBase + l16) * 64;
  for (int kb = 0; kb < 16; ++kb) {
    const int k0 = kb * 4 + half * 2;
    v2f a = *(const v2f*)(arow + k0);
    v2f b = *(const v2f*)&sB[kb * 64 + l16 * 4 + half * 2];
    acc = wmma_f32(a, b, acc);
  }
  const int drow0 = rowBase + half * 8;
#pragma unroll
  for (int i = 0; i < 8; ++i)
    O[(size_t)(drow0 + i) * 16 + l16] = acc[i];
}

__global__ void k_headmean(const float* __restrict__ tmp, const float* __restrict__ ob,
                           float* __restrict__ out) {
  int i = blockIdx.x * 256 + threadIdx.x;  // exactly N*16 = 800000 = 3125*256
  int n = i >> 4, c = i & 15;
  float s = 0.f;
#pragma unroll
  for (int t = 0; t < 8; ++t) s += tmp[(size_t)(n * 8 + t) * 16 + c];
  out[i] = s * 0.125f + ob[c];
}

// ---------------------------------------------------------------------------
extern "C" void kernel_launch(void* const* d_in, const int* in_sizes, int n_in,
                              void* d_out, int out_size, void* d_ws, size_t ws_size,
                              hipStream_t stream) {
  (void)in_sizes; (void)n_in; (void)out_size; (void)ws_size;

  const float* x = (const float*)d_in[0];
  const long long* ei = (const long long*)d_in[1];
  const float* ew = (const float*)d_in[2];
  const float* l0_tw = (const float*)d_in[3];
  const float* l0_tb = (const float*)d_in[4];
  const float* l0_g = (const float*)d_in[5];
  const float* l0_b = (const float*)d_in[6];
  const float* l0_gw = (const float*)d_in[7];
  const float* l0_gb = (const float*)d_in[8];
  const float* l1_tw = (const float*)d_in[9];
  const float* l1_tb = (const float*)d_in[10];
  const float* l1_g = (const float*)d_in[11];
  const float* l1_b = (const float*)d_in[12];
  const float* l1_gw = (const float*)d_in[13];
  const float* l1_gb = (const float*)d_in[14];
  const float* out_w = (const float*)d_in[15];
  const float* out_b = (const float*)d_in[16];
  float* out = (float*)d_out;

  // workspace carve-out (256B aligned)
  char* ws = (char*)d_ws;
  size_t off = 0;
  auto carve = [&](size_t bytes) {
    char* p = ws + off;
    off = (off + bytes + 255) & ~(size_t)255;
    return p;
  };
  const size_t big = (size_t)NN * TT * 64 * sizeof(float);  // 102.4 MB
  float* hA   = (float*)carve(big);
  float* hw   = (float*)carve(big);
  float* agg  = (float*)carve(big);
  float* wsm  = (float*)carve((size_t)EE * 4);
  float* nrm  = (float*)carve((size_t)EE * 4);
  float* deg  = (float*)carve((size_t)NN * 4);
  float* dinv = (float*)carve((size_t)NN * 4);
  float* sfn  = (float*)carve((size_t)NN * 4);
  int*   mxb  = (int*)carve(256);
  float* smv  = (float*)carve(256);

  const int GE = (EE + 255) / 256;            // 2344
  const int GN = (NN + 255) / 256;            // 196
  const int GEL4 = (NN * TT * 64 / 4) / 256;  // 25000 (exact)
  const int GR = (NN * TT) / 128;             // 3125 (exact)

  // --- edge softmax + symmetric GCN normalization (shared by both layers) ---
  k_init_scalar<<<1, 32, 0, stream>>>(mxb, smv);
  k_init_deg<<<GN, 256, 0, stream>>>(deg);
  k_edge_max<<<GE, 256, 0, stream>>>(ew, mxb);
  k_edge_expsum<<<GE, 256, 0, stream>>>(ew, mxb, smv);
  k_edge_wsm<<<GE, 256, 0, stream>>>(ew, ei, mxb, smv, wsm, deg);
  k_dinv<<<GN, 256, 0, stream>>>(deg, dinv, sfn);
  k_norm<<<GE, 256, 0, stream>>>(ei, wsm, dinv, nrm);

  // --- layer 0 ---
  k_conv_in<32, true><<<GR, 256, 0, stream>>>(x, l0_tw, l0_tb, l0_g, l0_b, hA);
  k_gemm64<<<GR, 256, 0, stream>>>(hA, l0_gw, hw);
  k_selfloop<<<GEL4, 256, 0, stream>>>((const v4f*)hw, sfn, (v4f*)agg);
  k_scatter<<<EE, 256, 0, stream>>>(hw, ei, nrm, agg);
  k_finalize<<<GEL4, 256, 0, stream>>>((v4f*)agg, (const v4f*)l0_gb);

  // --- layer 1 (input = agg, layout [N][T][64]) ---
  k_conv_in<64, false><<<GR, 256, 0, stream>>>(agg, l1_tw, l1_tb, l1_g, l1_b, hA);
  k_gemm64<<<GR, 256, 0, stream>>>(hA, l1_gw, hw);
  k_selfloop<<<GEL4, 256, 0, stream>>>((const v4f*)hw, sfn, (v4f*)agg);
  k_scatter<<<EE, 256, 0, stream>>>(hw, ei, nrm, agg);
  k_finalize<<<GEL4, 256, 0, stream>>>((v4f*)agg, (const v4f*)l1_gb);

  // --- head: per-timestep linear, mean over T ---
  k_headgemm<<<GR, 256, 0, stream>>>(agg, out_w, hw);  // reuse hw as [N*T,16] tmp
  k_headmean<<<(NN * 16) / 256, 256, 0, stream>>>(hw, out_b, out);
}